// Sparsify1D_58548994179832
// MI455X (gfx1250) — compile-verified
//
#include <hip/hip_runtime.h>
#include <stdint.h>

// Sparsify1D: per-row exact k-th-largest threshold (k = D/2) + renormalize.
// Persistent blocks: 8 rows/block, ping-pong LDS buffers fed by
// global_load_async_to_lds_b128 (ASYNCcnt pipelined, non-temporal),
// per-wave radix histograms, wave32 shfl suffix-scan selection,
// non-temporal streaming output stores.

#define D_DIM    4096
#define K_SEL    2048u
#define NTHREADS 256
#define NWAVES   (NTHREADS / 32)
#define ROWS_PER_BLOCK 8
#define VPT      4      // float4 loads per thread per row (16 elems)

typedef __attribute__((ext_vector_type(4))) float f32x4;

__device__ __forceinline__ unsigned f32_to_ordkey(float f) {
  unsigned u = __float_as_uint(f);
  // monotonic: larger float <-> larger unsigned key
  return (u & 0x80000000u) ? ~u : (u | 0x80000000u);
}

__global__ __launch_bounds__(NTHREADS) void sparsify_rows(
    const float* __restrict__ x, float* __restrict__ out) {
  __shared__ __align__(16) float s_x[2][D_DIM];   // 32 KB ping-pong row buffers
  __shared__ unsigned s_whist[NWAVES * 256];      // 8 KB per-wave histograms
  __shared__ unsigned s_tot[256];
  __shared__ unsigned s_prefix;
  __shared__ unsigned s_kk;
  __shared__ float    s_sum;

  const int tid  = threadIdx.x;
  const int wave = tid >> 5;
  const size_t row0 = (size_t)blockIdx.x * ROWS_PER_BLOCK;

  const uint32_t toff = (uint32_t)tid * 64u;      // 16 floats per thread
  // low 32 bits of a flat LDS address == wave-relative LDS byte offset
  const uint32_t lb0 = (uint32_t)(uintptr_t)(&s_x[0][0]);
  const uint32_t lb1 = (uint32_t)(uintptr_t)(&s_x[1][0]);

  // prologue: async copy row0 -> buffer 0 (single-touch data: NT hint)
  {
    const float* rp = x + row0 * D_DIM;
#pragma unroll
    for (int j = 0; j < VPT; ++j) {
      const uint32_t boff = toff + (uint32_t)j * 16u;
      asm volatile("global_load_async_to_lds_b128 %0, %1, %2 th:TH_LOAD_NT"
                   :: "v"(lb0 + boff), "v"(boff), "s"(rp) : "memory");
    }
  }

  for (int r = 0; r < ROWS_PER_BLOCK; ++r) {
    const int cur = r & 1;
    const uint32_t lbn = (cur ? lb0 : lb1);

    if (r + 1 < ROWS_PER_BLOCK) {
      // prefetch next row into the other buffer, then retire current row's
      // 4 older transfers (async loads complete in order per wave)
      const float* rp = x + (row0 + r + 1) * D_DIM;
#pragma unroll
      for (int j = 0; j < VPT; ++j) {
        const uint32_t boff = toff + (uint32_t)j * 16u;
        asm volatile("global_load_async_to_lds_b128 %0, %1, %2 th:TH_LOAD_NT"
                     :: "v"(lbn + boff), "v"(boff), "s"(rp) : "memory");
      }
      asm volatile("s_wait_asynccnt 4" ::: "memory");
    } else {
      asm volatile("s_wait_asynccnt 0" ::: "memory");
    }

    if (tid == 0) { s_prefix = 0u; s_kk = K_SEL; s_sum = 0.0f; }
    __syncthreads();   // row data + scalars visible block-wide

    const f32x4* sx4 = (const f32x4*)&s_x[cur][0];

    // ---- 4-pass radix select (MSB->LSB), per-wave private histograms ----
    for (int shift = 24; shift >= 0; shift -= 8) {
#pragma unroll
      for (int i = 0; i < NWAVES; ++i) s_whist[i * 256 + tid] = 0u;
      __syncthreads();
      const unsigned prefix = s_prefix;
      const unsigned himask = (shift == 24) ? 0u : (0xFFFFFFFFu << (shift + 8));
#pragma unroll
      for (int j = 0; j < VPT; ++j) {
        const f32x4 v = sx4[tid * VPT + j];
#pragma unroll
        for (int e = 0; e < 4; ++e) {
          const unsigned u = f32_to_ordkey(v[e]);
          if ((u & himask) == prefix)
            atomicAdd(&s_whist[wave * 256 + ((u >> shift) & 0xFFu)], 1u);
        }
      }
      __syncthreads();
      // merge the 8 wave histograms (bin 'tid')
      unsigned tot = 0;
#pragma unroll
      for (int i = 0; i < NWAVES; ++i) tot += s_whist[i * 256 + tid];
      s_tot[tid] = tot;
      __syncthreads();
      // wave 0: suffix scan over 256 bins (8 bins/lane + shfl_down log-scan)
      if (tid < 32) {
        unsigned t[8], lsum = 0;
#pragma unroll
        for (int i = 0; i < 8; ++i) { t[i] = s_tot[tid * 8 + i]; lsum += t[i]; }
        unsigned incl = lsum;
#pragma unroll
        for (int off = 1; off < 32; off <<= 1) {
          const unsigned o = __shfl_down(incl, off, 32);
          if (tid + off < 32) incl += o;
        }
        unsigned above = incl - lsum;      // counts in all higher lanes' bins
        const unsigned kk = s_kk;
#pragma unroll
        for (int i = 7; i >= 0; --i) {     // high bin -> low bin within lane
          const unsigned c = t[i];
          if (above < kk && above + c >= kk) {   // exactly one (lane,i) hits
            s_prefix = prefix | ((unsigned)(tid * 8 + i) << shift);
            s_kk = kk - above;
          }
          above += c;
        }
      }
      __syncthreads();
    }

    // exact key of the k-th largest -> threshold float (== top_k[:, -1])
    const unsigned pk = s_prefix;
    const unsigned tb = (pk & 0x80000000u) ? (pk & 0x7FFFFFFFu) : ~pk;
    const float thresh = __uint_as_float(tb);

    // ---- sum of kept values: wave32 shfl reduce + ds_add_f32 ----
    float local = 0.0f;
#pragma unroll
    for (int j = 0; j < VPT; ++j) {
      const f32x4 v = sx4[tid * VPT + j];
#pragma unroll
      for (int e = 0; e < 4; ++e)
        if (v[e] >= thresh) local += v[e];
    }
#pragma unroll
    for (int off = 16; off > 0; off >>= 1)
      local += __shfl_xor(local, off, 32);
    if ((tid & 31) == 0) atomicAdd(&s_sum, local);
    __syncthreads();

    // out = kept * D / sum   (== res / (sum/D)); streaming NT stores
    const float scale = (float)D_DIM / s_sum;
    f32x4* __restrict__ outv = (f32x4*)(out + (row0 + r) * D_DIM);
#pragma unroll
    for (int j = 0; j < VPT; ++j) {
      const f32x4 v = sx4[tid * VPT + j];
      f32x4 rr;
#pragma unroll
      for (int e = 0; e < 4; ++e)
        rr[e] = (v[e] >= thresh) ? v[e] * scale : 0.0f;
      __builtin_nontemporal_store(rr, &outv[tid * VPT + j]);
    }
    __syncthreads();   // all reads of s_x[cur] done before async overwrites it
  }
}

extern "C" void kernel_launch(void* const* d_in, const int* in_sizes, int n_in,
                              void* d_out, int out_size, void* d_ws, size_t ws_size,
                              hipStream_t stream) {
  (void)n_in; (void)out_size; (void)d_ws; (void)ws_size;
  const float* x = (const float*)d_in[0];
  float* out = (float*)d_out;
  const int nrows   = in_sizes[0] / D_DIM;        // 16384
  const int nblocks = nrows / ROWS_PER_BLOCK;     // 2048
  sparsify_rows<<<dim3(nblocks), dim3(NTHREADS), 0, stream>>>(x, out);
}